// _DecoderZX_5179730559212
// MI455X (gfx1250) — compile-verified
//
#include <hip/hip_runtime.h>
#include <hip/hip_bf16.h>

// Problem constants (fixed by the reference)
#define B_SAMP 1024
#define N_IN   64
#define N_OUT  4096
#define N_BATCH 64
#define N_CONT 16

// LDS row strides (padded to avoid bank conflicts)
#define HSTRIDE 4100   // 16 rows of h  (16*4100*4  = 262,400 B)
#define ZSTRIDE 68     // 16 rows of z  (16*68*4    =   4,352 B)
#define CSTRIDE 20     // 16 rows of cc (16*20*4    =   1,280 B)

typedef __attribute__((ext_vector_type(2))) float v2f;
typedef __attribute__((ext_vector_type(8))) float v8f;

static __device__ __forceinline__ v8f wmma4(v2f a, v2f b, v8f c) {
    // D = A(16x4,f32) * B(4x16,f32) + C(16x16,f32)
    return __builtin_amdgcn_wmma_f32_16x16x4_f32(
        /*neg_a=*/false, a, /*neg_b=*/false, b,
        /*c_mod=*/(short)0, c, /*reuse_a=*/false, /*reuse_b=*/false);
}

// ---------------------------------------------------------------------------
// Kernel 1: counting-sort samples by batch index into <=128 batch-pure tiles
// of 16 sample slots each (slot = -1 means padded/empty).
// ---------------------------------------------------------------------------
__global__ __launch_bounds__(1024) void prep_tiles_kernel(
    const int* __restrict__ idx, int* __restrict__ tiles_batch,
    int* __restrict__ slots)
{
    __shared__ int hist[N_BATCH];
    __shared__ int cursor[N_BATCH];
    __shared__ int sbase[N_BATCH];
    int t = threadIdx.x;            // 1024 threads == B_SAMP
    if (t < N_BATCH) { hist[t] = 0; cursor[t] = 0; }
    __syncthreads();
    int b = idx[t];
    atomicAdd(&hist[b], 1);
    __syncthreads();
    if (t == 0) {
        int tc = 0;
        for (int bb = 0; bb < N_BATCH; ++bb) {
            sbase[bb] = tc * 16;
            int nt = (hist[bb] + 15) >> 4;
            for (int k = 0; k < nt; ++k) tiles_batch[tc++] = bb;
        }
        for (; tc < 128; ++tc) tiles_batch[tc] = -1;
    }
    __syncthreads();
    slots[t] = -1;
    slots[t + 1024] = -1;           // 128*16 = 2048 slots total
    __syncthreads();
    int p = atomicAdd(&cursor[b], 1);
    slots[sbase[b] + p] = t;
}

// ---------------------------------------------------------------------------
// Kernel 2: fused  h = z@W + A_b@z + h3_b + cc@cW  -> softmax -> *size_factor
// One block = one 16-sample batch-pure tile x all 4096 outputs (h in LDS).
// ---------------------------------------------------------------------------
__global__ __launch_bounds__(256) void decoder_main_kernel(
    const float* __restrict__ z, const float* __restrict__ sizef,
    const float* __restrict__ cont, const float* __restrict__ amat_W,
    const float* __restrict__ embed_A, const float* __restrict__ embed_h3,
    const float* __restrict__ cont_W, const int* __restrict__ tiles_batch,
    const int* __restrict__ slots, float* __restrict__ out)
{
    extern __shared__ float smem[];
    float* hbuf  = smem;                        // [16][HSTRIDE]
    float* ztile = hbuf  + 16 * HSTRIDE;        // [16][ZSTRIDE]
    float* ctile = ztile + 16 * ZSTRIDE;        // [16][CSTRIDE]
    float* sfbuf = ctile + 16 * CSTRIDE;        // [16]
    int*   sids  = (int*)(sfbuf + 16);          // [16]

    const int tile = blockIdx.x;
    const int b = tiles_batch[tile];
    if (b < 0) return;                          // uniform per block
    const int t = threadIdx.x;

    if (t < 16) {
        int s = slots[tile * 16 + t];
        sids[t] = s;
        sfbuf[t] = (s >= 0) ? sizef[s] : 1.0f;
    }
    __syncthreads();

    // Stage z tile (16x64) and continuous-covariate tile (16x16); pad rows = 0
    for (int i = t; i < 16 * N_IN; i += 256) {
        int m = i >> 6, l = i & 63;
        int s = sids[m];
        ztile[m * ZSTRIDE + l] = (s >= 0) ? z[s * N_IN + l] : 0.0f;
    }
    for (int i = t; i < 16 * N_CONT; i += 256) {
        int m = i >> 4, l = i & 15;
        int s = sids[m];
        ctile[m * CSTRIDE + l] = (s >= 0) ? cont[s * N_CONT + l] : 0.0f;
    }
    __syncthreads();

    const int wave = t >> 5;
    const int lane = t & 31;
    const int half = lane >> 4;                 // wave half
    const int ln   = lane & 15;                 // A: row M; B/C: col N

    // Preload all A-fragments for the K=64 z GEMMs (16 chunks of K=4)
    // ISA layout: lanes0-15 -> {K=0,K=1}, lanes16-31 -> {K=2,K=3}, M=lane%16
    v2f afrag[16];
#pragma unroll
    for (int c = 0; c < 16; ++c) {
        const float* p = &ztile[ln * ZSTRIDE + 4 * c + 2 * half];
        v2f v; v.x = p[0]; v.y = p[1];
        afrag[c] = v;
    }
    v2f cfrag[4];
#pragma unroll
    for (int c = 0; c < 4; ++c) {
        const float* p = &ctile[ln * CSTRIDE + 4 * c + 2 * half];
        v2f v; v.x = p[0]; v.y = p[1];
        cfrag[c] = v;
    }

    const float* Eb  = embed_A  + ((size_t)b << 18);      // b * N_OUT*N_IN
    const float* h3b = embed_h3 + (size_t)b * N_OUT;

    // Each wave owns column tiles ct = wave, wave+8, ... (32 tiles of 16 cols)
    for (int ct = wave; ct < N_OUT / 16; ct += 8) {
        const int g0 = ct * 16;
        const int gcol = g0 + ln;
        const float* ep = Eb + (size_t)gcol * N_IN;
        if (ct + 8 < N_OUT / 16)
            __builtin_prefetch(ep + 128 * N_IN, 0, 1);    // next tile's A_b rows

        v8f acc = {};
        // h1 : z @ amat_W   (B frag strided: W[k][g])
#pragma unroll
        for (int c = 0; c < 16; ++c) {
            int k = 4 * c + 2 * half;
            v2f bf;
            bf.x = amat_W[(size_t)k * N_OUT + gcol];
            bf.y = amat_W[(size_t)(k + 1) * N_OUT + gcol];
            acc = wmma4(afrag[c], bf, acc);
        }
        // h2 : z @ A_b^T    (B frag contiguous: E[b, g*64 + k] -> float2)
#pragma unroll
        for (int c = 0; c < 16; ++c) {
            int k = 4 * c + 2 * half;
            v2f bf = *(const v2f*)(ep + k);
            acc = wmma4(afrag[c], bf, acc);
        }
        // cc @ cont_W (K=16 -> 4 chunks)
#pragma unroll
        for (int c = 0; c < 4; ++c) {
            int k = 4 * c + 2 * half;
            v2f bf;
            bf.x = cont_W[(size_t)k * N_OUT + gcol];
            bf.y = cont_W[(size_t)(k + 1) * N_OUT + gcol];
            acc = wmma4(cfrag[c], bf, acc);
        }
        const float h3v = h3b[gcol];
        // C layout: VGPR r -> M = r + 8*half, N = ln
#pragma unroll
        for (int r = 0; r < 8; ++r) {
            int m = r + 8 * half;
            hbuf[m * HSTRIDE + gcol] = acc[r] + h3v;
        }
    }
    __syncthreads();

    // Row softmax: wave w handles rows 2w (lanes 0-15) and 2w+1 (lanes 16-31)
    const int row = 2 * wave + half;
    float* hrow = hbuf + row * HSTRIDE;

    float mx = -3.402823466e38f;
    for (int c = ln; c < N_OUT; c += 16) mx = fmaxf(mx, hrow[c]);
#pragma unroll
    for (int off = 1; off < 16; off <<= 1) mx = fmaxf(mx, __shfl_xor(mx, off, 32));

    float sum = 0.0f;
    for (int c = ln; c < N_OUT; c += 16) {
        float e = __expf(hrow[c] - mx);
        hrow[c] = e;
        sum += e;
    }
#pragma unroll
    for (int off = 1; off < 16; off <<= 1) sum += __shfl_xor(sum, off, 32);

    const int s = sids[row];
    if (s >= 0) {
        const float scale = sfbuf[row] / sum;
        float* orow = out + (size_t)s * N_OUT;
        for (int c = ln; c < N_OUT; c += 16) orow[c] = hrow[c] * scale;
    }
}

// ---------------------------------------------------------------------------
// Kernel 3: inverse_dispersion = exp(px_r)
// ---------------------------------------------------------------------------
__global__ void exp_kernel(const float* __restrict__ px_r, float* __restrict__ o)
{
    int i = blockIdx.x * blockDim.x + threadIdx.x;
    if (i < N_OUT) o[i] = __expf(px_r[i]);
}

extern "C" void kernel_launch(void* const* d_in, const int* in_sizes, int n_in,
                              void* d_out, int out_size, void* d_ws, size_t ws_size,
                              hipStream_t stream)
{
    const float* z    = (const float*)d_in[0];
    const int*   bc   = (const int*)  d_in[1];
    const float* sf   = (const float*)d_in[2];
    const float* cc   = (const float*)d_in[3];
    const float* aW   = (const float*)d_in[4];
    const float* eA   = (const float*)d_in[5];
    const float* eh3  = (const float*)d_in[6];
    const float* cW   = (const float*)d_in[7];
    const float* pxr  = (const float*)d_in[8];
    float* out = (float*)d_out;

    int* ws = (int*)d_ws;
    int* tiles_batch = ws;          // 128 ints
    int* slots       = ws + 128;    // 2048 ints

    prep_tiles_kernel<<<1, 1024, 0, stream>>>(bc, tiles_batch, slots);

    size_t smem = (size_t)(16 * HSTRIDE + 16 * ZSTRIDE + 16 * CSTRIDE + 16) * sizeof(float)
                + 16 * sizeof(int);   // ~268 KB of the 320 KB WGP LDS
    decoder_main_kernel<<<128, 256, smem, stream>>>(
        z, sf, cc, aW, eA, eh3, cW, tiles_batch, slots, out);

    exp_kernel<<<16, 256, 0, stream>>>(pxr, out + (size_t)B_SAMP * N_OUT);
}